// DurationLoss_46961172414631
// MI455X (gfx1250) — compile-verified
//
#include <hip/hip_runtime.h>
#include <math.h>

#ifndef __has_builtin
#define __has_builtin(x) 0
#endif

#if __has_builtin(__builtin_amdgcn_global_load_async_to_lds_b128)
#define USE_ASYNC_LDS 1
#else
#define USE_ASYNC_LDS 0
#endif

#if __has_builtin(__builtin_amdgcn_s_wait_asynccnt)
#define WAIT_ASYNC(n) __builtin_amdgcn_s_wait_asynccnt(n)
#else
#define WAIT_ASYNC(n) asm volatile("s_wait_asynccnt %0" ::"i"(n) : "memory")
#endif

#define B_ 1024
#define T_ 8192
#define NW 2048     // NUM_WORDS
#define TS 1024     // tile size (elements)
#define NT 256      // threads per block (8 wave32)
#define EPT 4       // elements per thread per tile (16B = one b128)
#define NTILES (T_ / TS)

typedef int v4i __attribute__((vector_size(16)));
typedef v4i __attribute__((address_space(1))) v4i_glb;
typedef v4i __attribute__((address_space(3))) v4i_lds;

// 16-byte async copy global -> LDS (one lane = one b128 transfer)
__device__ __forceinline__ void async_copy16(const void* gsrc, void* ldst) {
#if USE_ASYNC_LDS
  __builtin_amdgcn_global_load_async_to_lds_b128(
      (v4i_glb*)gsrc, (v4i_lds*)ldst, 0, 0);
#else
  *reinterpret_cast<float4*>(ldst) = *reinterpret_cast<const float4*>(gsrc);
#endif
}

// rules gaps at one position: g1 (init-token gap), g2 (ratio gap), f2 (ratio fired)
__device__ __forceinline__ void rule_gaps(float dp, float dpn, int tok,
                                          bool valid_next,
                                          float& g1, float& g2, bool& f2) {
  bool ii = false;
  float ex = 0.f;
  if (tok == 94 || tok == 100 || tok == 92) { ii = true; ex = 2.f; }
  else if (tok == 122)                      { ii = true; ex = 3.f; }
  else if (tok == 43 || tok == 27)          { ii = true; ex = 5.f; }
  float gap1 = dp - ex;
  g1 = (ii && gap1 > 0.f) ? gap1 : 0.f;
  bool ir = (tok == 44 || tok == 28 || tok == 29 || tok == 27 || tok == 121 || tok == 43);
  f2 = ir && valid_next && (3.f * dp > dpn);
  g2 = f2 ? (dp - dpn / 3.f) : 0.f;
}

__global__ __launch_bounds__(NT) void dur_loss_main(
    const float* __restrict__ dp_g, const float* __restrict__ gt_g,
    const int* __restrict__ w_g, const int* __restrict__ tok_g,
    float* __restrict__ part) {
  __shared__ __align__(16) float s_wp[NW];
  __shared__ __align__(16) float s_wg[NW];
  __shared__ __align__(16) float s_dp[2][TS + 4];  // +2 halo, padded to 16B stride
  __shared__ __align__(16) int   s_tk[2][TS + 4];  // +1 halo
  __shared__ __align__(16) float s_gt[2][TS];
  __shared__ __align__(16) int   s_w[2][TS];
  __shared__ float s_red[NT];

  const int b = blockIdx.x;
  const int tid = threadIdx.x;
  const long row = (long)b * T_;
  const int l0 = tid * EPT;

  for (int i = tid; i < NW; i += NT) { s_wp[i] = 0.f; s_wg[i] = 0.f; }

  auto issue = [&](int k) {
    const int buf = k & 1;
    const long base = row + (long)k * TS + l0;
    async_copy16(dp_g + base, &s_dp[buf][l0]);
    async_copy16(gt_g + base, &s_gt[buf][l0]);
    async_copy16(w_g + base, &s_w[buf][l0]);
    async_copy16(tok_g + base, &s_tk[buf][l0]);
    if (tid == 0) {  // halo: next-tile head (or past-row padding)
      const int t1 = k * TS + TS, t2 = t1 + 1;
      s_dp[buf][TS]     = (t1 < T_) ? dp_g[row + t1] : 0.f;
      s_dp[buf][TS + 1] = (t2 < T_) ? dp_g[row + t2] : 0.f;
      s_tk[buf][TS]     = (t1 < T_) ? tok_g[row + t1] : -1;  // -1 => no rule fires
    }
  };

  float accp = 0.f, accr = 0.f, sdp = 0.f, sgt = 0.f;

  issue(0);
  for (int k = 0; k < NTILES; ++k) {
    const int buf = k & 1;
    if (k + 1 < NTILES) { issue(k + 1); WAIT_ASYNC(4); }  // keep newest 4 in flight
    else { WAIT_ASYNC(0); }
    __syncthreads();

    const int ts = k * TS;
#pragma unroll
    for (int j = 0; j < EPT; ++j) {
      const int l = l0 + j;
      const int t = ts + l;
      const float dp = s_dp[buf][l];
      const float dpn1 = s_dp[buf][l + 1];
      const float dpn2 = s_dp[buf][l + 2];
      const int tok = s_tk[buf][l];
      const int tokn = s_tk[buf][l + 1];
      const float gt = s_gt[buf][l];
      const int w = s_w[buf][l];

      float g1, g2, g1n, g2n;
      bool f2, f2n;
      rule_gaps(dp, dpn1, tok, t < T_ - 1, g1, g2, f2);
      rule_gaps(dpn1, dpn2, tokn, (t + 1) < T_ - 1, g1n, g2n, f2n);
      const float self_adj = f2 ? g2 : g1;
      const float dr = dp - self_adj + g1n + g2n;

      const float lp = logf(dp + 1.f);
      float d;
      d = lp - logf(gt + 1.f); accp += d * d;  // pdur SSE
      d = lp - logf(dr + 1.f); accr += d * d;  // rules SSE

      const float dpc = fmaxf(dp, 0.f);
      sdp += dpc;
      sgt += gt;
      atomicAdd(&s_wp[w], dpc);  // ds_add_f32 segment sums
      atomicAdd(&s_wg[w], gt);
    }
    __syncthreads();
  }

  // word-level SSE (skip word 0)
  float accw = 0.f;
  for (int w = 1 + tid; w < NW; w += NT) {
    float d = logf(s_wp[w] + 1.f) - logf(s_wg[w] + 1.f);
    accw += d * d;
  }

  auto reduce = [&](float v) -> float {
    s_red[tid] = v;
    __syncthreads();
    for (int s = NT / 2; s > 0; s >>= 1) {
      if (tid < s) s_red[tid] += s_red[tid + s];
      __syncthreads();
    }
    float r = s_red[0];
    __syncthreads();
    return r;
  };

  accp = reduce(accp);
  accr = reduce(accr);
  accw = reduce(accw);
  sdp = reduce(sdp);
  sgt = reduce(sgt);

  if (tid == 0) {
    part[0 * B_ + b] = accp;
    part[1 * B_ + b] = accr;
    part[2 * B_ + b] = accw;
    const float ds = logf(sdp + 1.f) - logf(sgt + 1.f);
    part[3 * B_ + b] = ds * ds;
  }
}

__global__ __launch_bounds__(NT) void dur_loss_final(const float* __restrict__ part,
                                                     float* __restrict__ out) {
  __shared__ double s_red[NT];
  const int tid = threadIdx.x;
  double a0 = 0, a1 = 0, a2 = 0, a3 = 0;
  for (int i = tid; i < B_; i += NT) {
    a0 += (double)part[0 * B_ + i];
    a1 += (double)part[1 * B_ + i];
    a2 += (double)part[2 * B_ + i];
    a3 += (double)part[3 * B_ + i];
  }
  auto reduce = [&](double v) -> double {
    s_red[tid] = v;
    __syncthreads();
    for (int s = NT / 2; s > 0; s >>= 1) {
      if (tid < s) s_red[tid] += s_red[tid + s];
      __syncthreads();
    }
    double r = s_red[0];
    __syncthreads();
    return r;
  };
  a0 = reduce(a0);
  a1 = reduce(a1);
  a2 = reduce(a2);
  a3 = reduce(a3);
  if (tid == 0) {
    const double nPT = (double)B_ * (double)T_;
    const double loss = 0.6 * (a0 / nPT)                      // pdur
                      + 0.3 * (a1 / nPT)                      // rules
                      + 0.3 * (a2 / ((double)B_ * (NW - 1)))  // wdur
                      + 0.1 * (a3 / (double)B_);              // sdur
    out[0] = (float)loss;
  }
}

extern "C" void kernel_launch(void* const* d_in, const int* in_sizes, int n_in,
                              void* d_out, int out_size, void* d_ws, size_t ws_size,
                              hipStream_t stream) {
  const float* dur_pred = (const float*)d_in[0];
  const float* dur_gt = (const float*)d_in[1];
  const int* ph2word = (const int*)d_in[2];
  const int* txt_tokens = (const int*)d_in[3];
  float* part = (float*)d_ws;  // 4 * B_ floats = 16 KB

  dur_loss_main<<<B_, NT, 0, stream>>>(dur_pred, dur_gt, ph2word, txt_tokens, part);
  dur_loss_final<<<1, NT, 0, stream>>>(part, (float*)d_out);
}